// SoftNCutsLoss_19052474925476
// MI455X (gfx1250) — compile-verified
//
#include <hip/hip_runtime.h>

#define DD 16
#define LL 20
#define WW 20
#define NVOX 6400          // DD*LL*WW
#define ROW_TILES 400      // NVOX / 16
#define JSPLIT 4           // column-range split per row tile
#define JT_PER (ROW_TILES / JSPLIT)      // 100 column tiles per wave
#define NWAVES (ROW_TILES * JSPLIT)      // 1600 waves
#define INV9 (1.0f / 9.0f)               // 1/STD_INTENSITY^2

typedef float v2f __attribute__((ext_vector_type(2)));
typedef float v8f __attribute__((ext_vector_type(8)));

// ---------------------------------------------------------------------------
// Prep: per-voxel table {f, x, y, z} into workspace (L2-resident, ~102 KB)
// ---------------------------------------------------------------------------
__global__ __launch_bounds__(256) void prep_kernel(const float* __restrict__ patch,
                                                   float4* __restrict__ tbl) {
    int i = blockIdx.x * blockDim.x + threadIdx.x;
    if (i < NVOX) {
        int d   = i / (LL * WW);
        int rem = i - d * (LL * WW);
        int l   = rem / WW;
        int w   = rem - l * WW;
        tbl[i] = make_float4(patch[i], (float)d, (float)l, (float)w);
    }
}

// ---------------------------------------------------------------------------
// Main: each wave owns a 16-row tile of W and JT_PER column tiles.
// Generates W tiles in registers (fused single-exp affinity) and feeds them
// straight into V_WMMA_F32_16X16X4_F32 against B = [P | 1 | 0...] (16 cols).
// Accumulator C: cols 0..3 = (W@P), col 4 = row_sum(W).
// ---------------------------------------------------------------------------
__global__ __launch_bounds__(256) void ncut_wmma_kernel(const float4* __restrict__ tbl,
                                                        const float* __restrict__ prob,
                                                        float* __restrict__ part) {
    const int gtid = blockIdx.x * blockDim.x + threadIdx.x;
    const int wave = gtid >> 5;
    const int lane = threadIdx.x & 31;
    const int tile = wave >> 2;      // row tile 0..399
    const int seg  = wave & 3;       // column segment 0..3
    const int h    = lane >> 4;      // lane half (selects K pair in A/B layout)
    const int t    = lane & 15;      // A: row-in-tile; B/C: column index

    // Row-side data: A-matrix row M = lane%16 (both lane halves hold same row).
    const float4 r = tbl[tile * 16 + t];
    const float fi = r.x, xi = r.y, yi = r.z, zi = r.w;

    // B-column behavior per lane: cols 0..3 -> P[:,t], col 4 -> ones, else 0.
    const int   tc    = (t < 4) ? t : 0;
    const float bmask = (t < 4) ? 1.0f : 0.0f;
    const float bone  = (t == 4) ? 1.0f : 0.0f;

    v8f acc = {0.f, 0.f, 0.f, 0.f, 0.f, 0.f, 0.f, 0.f};

    const int j0 = seg * JT_PER;
    for (int j = j0; j < j0 + JT_PER; ++j) {
        const int base = j * 16 + h * 2;   // K = h*2 + {0,1} within each 16x4 chunk
#pragma unroll
        for (int c = 0; c < 4; ++c) {
            const int col0 = base + c * 4;
            const int col1 = col0 + 1;

            const float4 q0 = tbl[col0];
            const float4 q1 = tbl[col1];

            // Fused affinity: exp(-(df^2/9 + dx^2 + dy^2 + dz^2))
            float df0 = fi - q0.x, dx0 = xi - q0.y, dy0 = yi - q0.z, dz0 = zi - q0.w;
            float s0 = fmaf(df0 * df0, INV9, fmaf(dx0, dx0, fmaf(dy0, dy0, dz0 * dz0)));
            float df1 = fi - q1.x, dx1 = xi - q1.y, dy1 = yi - q1.z, dz1 = zi - q1.w;
            float s1 = fmaf(df1 * df1, INV9, fmaf(dx1, dx1, fmaf(dy1, dy1, dz1 * dz1)));
            float a0 = __expf(-s0);
            float a1 = __expf(-s1);

            float b0 = prob[col0 * 4 + tc] * bmask + bone;
            float b1 = prob[col1 * 4 + tc] * bmask + bone;

            v2f A = {a0, a1};
            v2f B = {b0, b1};
            // D(16x16,f32) += A(16x4 of W, f32) x B(4x16 of [P|1], f32)
            acc = __builtin_amdgcn_wmma_f32_16x16x4_f32(
                false, A, false, B, (short)0, acc, false, false);
        }
    }

    // Epilogue. C layout: lane holds col N = t; VGPR v holds row M = v + 8*h.
    float* wpart = part + wave * 16;
    if (t < 4) {
        // partial nums[t] = sum_rows P[row,t] * (W@P)[row,t]
        float s = 0.f;
#pragma unroll
        for (int v = 0; v < 8; ++v) {
            int row = tile * 16 + v + 8 * h;
            s += prob[row * 4 + t] * acc[v];
        }
        wpart[h * 4 + t] = s;               // slots 0..7
    } else if (t == 4) {
        // col 4 holds row_sum; partial dens[tt] = sum_rows P[row,tt] * row_sum[row]
        float d0 = 0.f, d1 = 0.f, d2 = 0.f, d3 = 0.f;
#pragma unroll
        for (int v = 0; v < 8; ++v) {
            int row = tile * 16 + v + 8 * h;
            float rs = acc[v];
            const float4 p = ((const float4*)prob)[row];
            d0 += p.x * rs; d1 += p.y * rs; d2 += p.z * rs; d3 += p.w * rs;
        }
        *(float4*)(wpart + 8 + h * 4) = make_float4(d0, d1, d2, d3);  // slots 8..15
    }
}

// ---------------------------------------------------------------------------
// Finalize: fixed-order (deterministic) reduction of per-wave partials.
// ---------------------------------------------------------------------------
__global__ void finalize_kernel(const float* __restrict__ part,
                                const int* __restrict__ kptr,
                                float* __restrict__ out) {
    __shared__ float s[8];
    int tid = threadIdx.x;
    if (tid < 8) {
        int t   = tid & 3;
        int off = (tid < 4) ? 0 : 8;    // nums slots 0..7, dens slots 8..15
        float acc = 0.f;
        for (int g = 0; g < NWAVES; ++g) {
            const float* wp = part + g * 16 + off;
            acc += wp[t] + wp[4 + t];
        }
        s[tid] = acc;
    }
    __syncthreads();
    if (tid == 0) {
        float loss = (float)(*kptr);
        for (int t = 0; t < 4; ++t) loss -= s[t] / s[4 + t];
        *out = loss;
    }
}

// ---------------------------------------------------------------------------
extern "C" void kernel_launch(void* const* d_in, const int* in_sizes, int n_in,
                              void* d_out, int out_size, void* d_ws, size_t ws_size,
                              hipStream_t stream) {
    const float* patch = (const float*)d_in[0];   // 6400 f32
    const float* prob  = (const float*)d_in[1];   // 6400*4 f32
    const int*   kptr  = (const int*)d_in[2];     // scalar k (=4)

    float4* tbl  = (float4*)d_ws;                         // 6400 * 16 B
    float*  part = (float*)d_ws + NVOX * 4;               // 1600 * 16 f32

    prep_kernel<<<(NVOX + 255) / 256, 256, 0, stream>>>(patch, tbl);
    ncut_wmma_kernel<<<(NWAVES * 32) / 256, 256, 0, stream>>>(tbl, prob, part);
    finalize_kernel<<<1, 32, 0, stream>>>(part, kptr, (float*)d_out);
}